// TopNPerAccuracy0_40699110097728
// MI455X (gfx1250) — compile-verified
//
#include <hip/hip_runtime.h>
#include <hip/hip_bf16.h>

typedef float v2f __attribute__((ext_vector_type(2)));
typedef float v8f __attribute__((ext_vector_type(8)));

#define NROWS 8192
#define KDIM  128
#define NPER  82          // int(8192 * 1 / 100) + 1
#define ROWS_PER_BLOCK 32 // two 16-row WMMA tiles per wave, shared B

// ---------------------------------------------------------------------------
// Kernel 1: per-row x2 inverse norms and scaled diagonal values.
//   rnorm2[j] = 1 / ||x2_j||
//   diag[i]   = dot(x1_i, x2_i) * rnorm2[i]
// Also zeroes the global hit counter (stream ordering makes this safe).
// ---------------------------------------------------------------------------
__global__ __launch_bounds__(256) void prep_kernel(const float* __restrict__ x1,
                                                   const float* __restrict__ x2,
                                                   float* __restrict__ rnorm2,
                                                   float* __restrict__ diag,
                                                   int* __restrict__ counter) {
    const int j = blockIdx.x * blockDim.x + threadIdx.x;
    if (j == 0) *counter = 0;
    if (j >= NROWS) return;
    const float4* a = (const float4*)(x1 + (size_t)j * KDIM);
    const float4* b = (const float4*)(x2 + (size_t)j * KDIM);
    float nn = 0.0f, dd = 0.0f;
#pragma unroll 8
    for (int k = 0; k < KDIM / 4; ++k) {
        const float4 av = a[k];
        const float4 bv = b[k];
        nn += bv.x * bv.x + bv.y * bv.y + bv.z * bv.z + bv.w * bv.w;
        dd += av.x * bv.x + av.y * bv.y + av.z * bv.z + av.w * bv.w;
    }
    const float rn = rsqrtf(nn);
    rnorm2[j] = rn;
    diag[j]   = dd * rn;
}

// ---------------------------------------------------------------------------
// Kernel 2: fused GEMM + rank counting.
// Block = 32 rows (two WMMA M-tiles per wave), 256 threads = 8 waves.
// Each wave owns j-tiles jt = wave, wave+8, ...; per tile it loads the B
// fragments ONCE into a register array (breaking load->wait->wmma
// serialization) and feeds two independent accumulator chains (rows 0-15
// and 16-31), halving B traffic per WMMA and doubling XDL-chain ILP.
//
// v_wmma_f32_16x16x4_f32 operand layout (ISA 7.12.2):
//   A (16x4):  lanes 0-15: M=lane, VGPR{0,1}=K{0,1}; lanes 16-31: K{2,3}
//   B (4x16):  lanes 0-15: N=lane, VGPR{0,1}=K{0,1}; lanes 16-31: K{2,3}
//   C (16x16): VGPR r: lanes 0-15 -> M=r, N=lane; lanes 16-31 -> M=8+r
// ---------------------------------------------------------------------------
__global__ __launch_bounds__(256) void rank_kernel(const float* __restrict__ x1,
                                                   const float* __restrict__ x2,
                                                   const float* __restrict__ rnorm2,
                                                   const float* __restrict__ diag,
                                                   int* __restrict__ counter) {
    __shared__ int cnt_lds[ROWS_PER_BLOCK];

    const int tid  = threadIdx.x;
    const int lane = tid & 31;
    const int wave = tid >> 5;      // 0..7
    const int n    = lane & 15;     // N column within tile (and A row M)
    const int half = lane >> 4;     // 0/1: selects K pair for A/B, M+8 for C
    const int row0 = blockIdx.x * ROWS_PER_BLOCK;

    if (tid < ROWS_PER_BLOCK) cnt_lds[tid] = 0;
    __syncthreads();

    // Preload both 16x128 A panels as 32 K-block fragments each (128 VGPRs).
    v2f afrag0[32], afrag1[32];
    {
        const float* arow0 = x1 + (size_t)(row0 + n) * KDIM + 2 * half;
        const float* arow1 = arow0 + (size_t)16 * KDIM;
#pragma unroll
        for (int kb = 0; kb < 32; ++kb) {
            afrag0[kb].x = arow0[4 * kb + 0];
            afrag0[kb].y = arow0[4 * kb + 1];
            afrag1[kb].x = arow1[4 * kb + 0];
            afrag1[kb].y = arow1[4 * kb + 1];
        }
    }

    // Diagonal thresholds for the 16 rows this lane produces (8 per tile).
    float dv0[8], dv1[8];
#pragma unroll
    for (int r = 0; r < 8; ++r) {
        dv0[r] = diag[row0 + r + 8 * half];
        dv1[r] = diag[row0 + 16 + r + 8 * half];
    }

    // Packed counters: low 16 bits = tile0 row r, high 16 bits = tile1 row r.
    // Max per-lane count is 64 tiles -> fits easily in 16 bits.
    int cnt[8];
#pragma unroll
    for (int r = 0; r < 8; ++r) cnt[r] = 0;

    // Wave-uniform loop bounds: EXEC stays all-ones around WMMA.
    for (int jt = wave; jt < NROWS / 16; jt += 8) {
        const int j0 = jt * 16;
        const float* brow = x2 + (size_t)(j0 + n) * KDIM + 2 * half;

        // Stage the whole B tile in registers first: 32 back-to-back b64
        // loads, so WMMAs below only pay incremental loadcnt waits.
        v2f bfrag[32];
#pragma unroll
        for (int kb = 0; kb < 32; ++kb) {
            bfrag[kb].x = brow[4 * kb + 0];
            bfrag[kb].y = brow[4 * kb + 1];
        }

        v8f c0 = {};
        v8f c1 = {};
#pragma unroll
        for (int kb = 0; kb < 32; ++kb) {
            // Two independent accumulation chains share each B fragment.
            c0 = __builtin_amdgcn_wmma_f32_16x16x4_f32(
                     false, afrag0[kb], false, bfrag[kb], (short)0, c0, false, false);
            c1 = __builtin_amdgcn_wmma_f32_16x16x4_f32(
                     false, afrag1[kb], false, bfrag[kb], (short)0, c1, false, false);
        }

        const int   j  = j0 + n;
        const float rn = rnorm2[j];
#pragma unroll
        for (int r = 0; r < 8; ++r) {
            const int i0 = row0 + r + 8 * half;
            const int i1 = i0 + 16;
            const float v0 = c0[r] * rn;
            const float v1 = c1[r] * rn;
            // stable-descending-sort rank contribution (skip diagonal itself)
            const bool g0 = (j != i0) && ((v0 > dv0[r]) || ((v0 == dv0[r]) && (j < i0)));
            const bool g1 = (j != i1) && ((v1 > dv1[r]) || ((v1 == dv1[r]) && (j < i1)));
            cnt[r] += (g0 ? 1 : 0) + (g1 ? 0x10000 : 0);
        }
    }

#pragma unroll
    for (int r = 0; r < 8; ++r) {
        const int c0n = cnt[r] & 0xFFFF;
        const int c1n = cnt[r] >> 16;
        if (c0n) atomicAdd(&cnt_lds[r + 8 * half], c0n);
        if (c1n) atomicAdd(&cnt_lds[16 + r + 8 * half], c1n);
    }
    __syncthreads();

    // rank_i < NPER  ->  row i is a hit
    if (tid < ROWS_PER_BLOCK && cnt_lds[tid] < NPER) atomicAdd(counter, 1);
}

// ---------------------------------------------------------------------------
// Kernel 3: scalar finalize. 8192 = 2^13, so hits/8192 is exact in fp32.
// ---------------------------------------------------------------------------
__global__ void finalize_kernel(const int* __restrict__ counter,
                                float* __restrict__ out) {
    if (threadIdx.x == 0 && blockIdx.x == 0)
        out[0] = (float)(*counter) * (1.0f / (float)NROWS);
}

extern "C" void kernel_launch(void* const* d_in, const int* in_sizes, int n_in,
                              void* d_out, int out_size, void* d_ws, size_t ws_size,
                              hipStream_t stream) {
    const float* x1 = (const float*)d_in[0];
    const float* x2 = (const float*)d_in[1];

    float* rnorm2  = (float*)d_ws;            // 8192 floats
    float* diag    = rnorm2 + NROWS;          // 8192 floats
    int*   counter = (int*)(diag + NROWS);    // 1 int

    prep_kernel<<<NROWS / 256, 256, 0, stream>>>(x1, x2, rnorm2, diag, counter);
    rank_kernel<<<NROWS / ROWS_PER_BLOCK, 256, 0, stream>>>(x1, x2, rnorm2, diag, counter);
    finalize_kernel<<<1, 1, 0, stream>>>(counter, (float*)d_out);
}